// MultiHeadedAttention_original_14199161881133
// MI455X (gfx1250) — compile-verified
//
#include <hip/hip_runtime.h>
#include <hip/hip_bf16.h>

// MultiHeadedAttention with relative position embeddings, fp32, CDNA5 WMMA.
// B=4, S=1024, D=1024, H=16, DK=DV=64, MAX_REL=32 (65 buckets).

typedef __attribute__((ext_vector_type(2))) float v2f;
typedef __attribute__((ext_vector_type(8))) float v8f;
typedef __attribute__((ext_vector_type(4))) int   i4;

#define S_LEN 1024
#define D_MOD 1024
#define NHEAD 16
#define HDIM  64
#define NREL  65
#define BH    64        // B*H
#define MROWS 4096      // B*S

__device__ __forceinline__ v8f wmma4(v2f a, v2f b, v8f c) {
  // D = A(16x4,f32) * B(4x16,f32) + C(16x16,f32)
  return __builtin_amdgcn_wmma_f32_16x16x4_f32(false, a, false, b, (short)0, c,
                                               false, false);
}

// ---- CDNA5 async global->LDS staging (ASYNCcnt path), with safe fallback ---
#if defined(__gfx1250__) && __has_builtin(__builtin_amdgcn_global_load_async_to_lds_b128)
#define HAVE_ASYNC_LDS 1
typedef __attribute__((address_space(1))) i4 gi4;   // global 16B vector
typedef __attribute__((address_space(3))) i4 li4;   // LDS 16B vector
#endif

__device__ __forceinline__ void cp16(const float* g, float* l) {
#ifdef HAVE_ASYNC_LDS
  __builtin_amdgcn_global_load_async_to_lds_b128((gi4*)g, (li4*)l, 0, 0);
#else
  *reinterpret_cast<float4*>(l) = *reinterpret_cast<const float4*>(g);
#endif
}

__device__ __forceinline__ void async_wait() {
#ifdef HAVE_ASYNC_LDS
#if __has_builtin(__builtin_amdgcn_s_wait_asynccnt)
  __builtin_amdgcn_s_wait_asynccnt(0);
#else
  asm volatile("s_wait_asynccnt 0" ::: "memory");
#endif
#endif
}

// ---------------------------------------------------------------------------
// Generic 4096x1024x1024 fp32 GEMM.  64(M) x 128(N) tile per 256-thread block;
// 8 waves in a 2x4 grid, each wave owns a 32x32 register tile (2x2 WMMA).
// Double-buffered async LDS staging over K chunks of 32.
// HEADSPLIT=1 -> out[b,h,s,d] with scale;  HEADSPLIT=0 -> row-major out[m,n].
// ---------------------------------------------------------------------------
template <int HEADSPLIT>
__global__ __launch_bounds__(256) void gemm_kernel(
    const float* __restrict__ X, const float* __restrict__ W,
    const float* __restrict__ bias, float* __restrict__ out, float scale) {
  __shared__ float As[2][64][36];    // [buf][m][k]   (36*4B = 144B rows, 16B ok)
  __shared__ float Bs[2][32][132];   // [buf][k][n]   (132*4B = 528B rows)
  const int tid = threadIdx.x;
  const int wid = tid >> 5, lane = tid & 31;
  const int lm = lane & 15, half = lane >> 4;
  const int m0 = blockIdx.y * 64;
  const int n0 = blockIdx.x * 128;
  const int WM = (wid & 1) * 32;
  const int WN = (wid >> 1) * 32;

  auto stage = [&](int buf, int kk) {
#pragma unroll
    for (int rnd = 0; rnd < 2; ++rnd) {        // A: 64x32 floats = 512 x 16B
      int idx = rnd * 256 + tid;
      int r = idx >> 3, c4 = (idx & 7) * 4;
      cp16(&X[(size_t)(m0 + r) * D_MOD + kk + c4], &As[buf][r][c4]);
    }
#pragma unroll
    for (int rnd = 0; rnd < 4; ++rnd) {        // B: 32x128 floats = 1024 x 16B
      int idx = rnd * 256 + tid;
      int r = idx >> 5, c4 = (idx & 31) * 4;
      cp16(&W[(size_t)(kk + r) * D_MOD + n0 + c4], &Bs[buf][r][c4]);
    }
  };

  v8f acc[2][2] = {};
  stage(0, 0);
  async_wait();
  __syncthreads();

  int cur = 0;
  for (int kk = 0; kk < D_MOD; kk += 32) {
    if (kk + 32 < D_MOD) stage(cur ^ 1, kk + 32);
#pragma unroll
    for (int k0 = 0; k0 < 32; k0 += 4) {
      v2f a[2], b[2];
#pragma unroll
      for (int mi = 0; mi < 2; ++mi) {
        a[mi].x = As[cur][WM + mi * 16 + lm][k0 + 2 * half];
        a[mi].y = As[cur][WM + mi * 16 + lm][k0 + 2 * half + 1];
      }
#pragma unroll
      for (int ni = 0; ni < 2; ++ni) {
        b[ni].x = Bs[cur][k0 + 2 * half][WN + ni * 16 + lm];
        b[ni].y = Bs[cur][k0 + 2 * half + 1][WN + ni * 16 + lm];
      }
#pragma unroll
      for (int mi = 0; mi < 2; ++mi)
#pragma unroll
        for (int ni = 0; ni < 2; ++ni)
          acc[mi][ni] = wmma4(a[mi], b[ni], acc[mi][ni]);
    }
    async_wait();
    __syncthreads();
    cur ^= 1;
  }

#pragma unroll
  for (int ni = 0; ni < 2; ++ni) {
    const int n = n0 + WN + ni * 16 + lm;
    const float bval = bias[n];
#pragma unroll
    for (int mi = 0; mi < 2; ++mi)
#pragma unroll
      for (int v = 0; v < 8; ++v) {
        int m = m0 + WM + mi * 16 + v + 8 * half;
        float val = (acc[mi][ni][v] + bval) * scale;
        if (HEADSPLIT) {
          int bb = m >> 10, s = m & 1023;   // m = b*S + s
          int h = n >> 6, d = n & 63;       // n = h*64 + d
          out[((((size_t)bb * NHEAD + h) << 10) + s) * HDIM + d] = val;
        } else {
          out[(size_t)m * D_MOD + n] = val;
        }
      }
  }
}

// ---------------------------------------------------------------------------
// QE[row, r] = dot(q[row, :], emb_k[r, :])   row in [0, B*H*S), r in [0,65)
// emb_k (16.6KB) stays hot in the 192MB L2.
// ---------------------------------------------------------------------------
__global__ __launch_bounds__(96) void qe_kernel(const float* __restrict__ q,
                                                const float* __restrict__ emb_k,
                                                float* __restrict__ qe) {
  __shared__ float qs[HDIM];
  const int row = blockIdx.x;
  const int tid = threadIdx.x;
  if (tid < HDIM) qs[tid] = q[(size_t)row * HDIM + tid];
  __syncthreads();
  if (tid < NREL) {
    const float* e = emb_k + (size_t)tid * HDIM;
    float acc = 0.f;
#pragma unroll
    for (int d = 0; d < HDIM; ++d) acc += qs[d] * e[d];
    qe[(size_t)row * NREL + tid] = acc;
  }
}

// ---------------------------------------------------------------------------
// Raw scores: attn[bh,i,j] = q_i . k_j + QE[bh,i, clip(j-i)+32]
// 64(i) x 128(j) tile; K=64 in one async LDS stage.  K tile kept row-major
// ([j][d]); B fragments read it with stride 68 (conflict-free).
// ---------------------------------------------------------------------------
__global__ __launch_bounds__(256) void scores_kernel(
    const float* __restrict__ q, const float* __restrict__ k,
    const float* __restrict__ qe, float* __restrict__ attn) {
  __shared__ float Qs[64][68];    // [i][d]
  __shared__ float Kl[128][68];   // [j][d]
  const int tid = threadIdx.x;
  const int wid = tid >> 5, lane = tid & 31;
  const int lm = lane & 15, half = lane >> 4;
  const int bh = blockIdx.z;
  const int m0 = blockIdx.y * 64;
  const int n0 = blockIdx.x * 128;
  const int WM = (wid & 1) * 32;
  const int WN = (wid >> 1) * 32;
  const float* qb = q + (size_t)bh * S_LEN * HDIM;
  const float* kb = k + (size_t)bh * S_LEN * HDIM;

#pragma unroll
  for (int rnd = 0; rnd < 4; ++rnd) {          // Q: 64x64 = 1024 x 16B
    int idx = rnd * 256 + tid;
    int r = idx >> 4, c4 = (idx & 15) * 4;
    cp16(&qb[(size_t)(m0 + r) * HDIM + c4], &Qs[r][c4]);
  }
#pragma unroll
  for (int rnd = 0; rnd < 8; ++rnd) {          // K: 128x64 = 2048 x 16B
    int idx = rnd * 256 + tid;
    int r = idx >> 4, c4 = (idx & 15) * 4;
    cp16(&kb[(size_t)(n0 + r) * HDIM + c4], &Kl[r][c4]);
  }
  async_wait();
  __syncthreads();

  v8f acc[2][2] = {};
#pragma unroll
  for (int k0 = 0; k0 < HDIM; k0 += 4) {
    v2f a[2], b[2];
#pragma unroll
    for (int mi = 0; mi < 2; ++mi) {
      a[mi].x = Qs[WM + mi * 16 + lm][k0 + 2 * half];
      a[mi].y = Qs[WM + mi * 16 + lm][k0 + 2 * half + 1];
    }
#pragma unroll
    for (int ni = 0; ni < 2; ++ni) {
      b[ni].x = Kl[WN + ni * 16 + lm][k0 + 2 * half];      // B[k,n] = k[n,k]
      b[ni].y = Kl[WN + ni * 16 + lm][k0 + 2 * half + 1];
    }
#pragma unroll
    for (int mi = 0; mi < 2; ++mi)
#pragma unroll
      for (int ni = 0; ni < 2; ++ni)
        acc[mi][ni] = wmma4(a[mi], b[ni], acc[mi][ni]);
  }

#pragma unroll
  for (int mi = 0; mi < 2; ++mi)
#pragma unroll
    for (int ni = 0; ni < 2; ++ni) {
      const int j = n0 + WN + ni * 16 + lm;
#pragma unroll
      for (int v = 0; v < 8; ++v) {
        int i = m0 + WM + mi * 16 + v + 8 * half;
        int rel = j - i;
        rel = (rel < -32) ? -32 : (rel > 32 ? 32 : rel);
        float val = acc[mi][ni][v] +
                    qe[((size_t)bh * S_LEN + i) * NREL + rel + 32];
        attn[((size_t)bh * S_LEN + i) * S_LEN + j] = val;
      }
    }
}

// ---------------------------------------------------------------------------
// Wave-per-row softmax (in place) + 65 bucket weights per row:
//   w[0]  = sum_{j<=i-32} a    w[64] = sum_{j>=i+32} a    w[1..63] = a[i+r-32]
// ---------------------------------------------------------------------------
__global__ __launch_bounds__(256) void softmax_kernel(float* __restrict__ attn,
                                                      float* __restrict__ wbuf) {
  __shared__ float rows[8][S_LEN];
  const int tid = threadIdx.x;
  const int wid = tid >> 5, lane = tid & 31;
  const int row = blockIdx.x * 8 + wid;    // [0, B*H*S)
  const int i = row & (S_LEN - 1);
  float* rp = attn + (size_t)row * S_LEN;

  float vals[32];
#pragma unroll
  for (int t = 0; t < 32; ++t) vals[t] = rp[t * 32 + lane];

  float m = -3.4e38f;
#pragma unroll
  for (int t = 0; t < 32; ++t) m = fmaxf(m, vals[t]);
#pragma unroll
  for (int off = 16; off; off >>= 1) m = fmaxf(m, __shfl_xor(m, off, 32));

  float s = 0.f;
#pragma unroll
  for (int t = 0; t < 32; ++t) { vals[t] = __expf(vals[t] - m); s += vals[t]; }
#pragma unroll
  for (int off = 16; off; off >>= 1) s += __shfl_xor(s, off, 32);
  const float inv = 1.0f / s;

  float slow = 0.f, shigh = 0.f;
#pragma unroll
  for (int t = 0; t < 32; ++t) {
    int j = t * 32 + lane;
    float a = vals[t] * inv;
    rp[j] = a;
    rows[wid][j] = a;
    if (j <= i - 32) slow += a;
    if (j >= i + 32) shigh += a;
  }
#pragma unroll
  for (int off = 16; off; off >>= 1) {
    slow += __shfl_xor(slow, off, 32);
    shigh += __shfl_xor(shigh, off, 32);
  }
  __syncthreads();

  float* wp = wbuf + (size_t)row * NREL;
  for (int idx = lane; idx < 63; idx += 32) {
    int r = idx + 1;              // rel = r-32 in [-31, 31]
    int j = i + r - 32;
    wp[r] = (j >= 0 && j < S_LEN) ? rows[wid][j] : 0.f;
  }
  if (lane == 0) { wp[0] = slow; wp[NREL - 1] = shigh; }
}

// ---------------------------------------------------------------------------
// ctx = attn @ V  (+ w @ emb_v epilogue), written in [B, S, H*64] layout.
// 128(i) x 64(d) tile per block, 8 waves 4x2, 32x32 per wave; Kt=32 double-
// buffered async staging.  The w-tile reuses the GEMM LDS via a union.
// ---------------------------------------------------------------------------
__global__ __launch_bounds__(256) void ctx_kernel(
    const float* __restrict__ attn, const float* __restrict__ v,
    const float* __restrict__ wbuf, const float* __restrict__ emb_v,
    float* __restrict__ ctx) {
  __shared__ union SM {
    struct { float A[2][128][36]; float B[2][32][68]; } g;  // 54272 B
    float W[128][66];                                       // 33792 B
  } sm;
  const int tid = threadIdx.x;
  const int wid = tid >> 5, lane = tid & 31;
  const int lm = lane & 15, half = lane >> 4;
  const int bh = blockIdx.y;
  const int m0 = blockIdx.x * 128;
  const int WM = (wid >> 1) * 32;
  const int WN = (wid & 1) * 32;
  const float* ab = attn + (size_t)bh * S_LEN * S_LEN;
  const float* vb = v + (size_t)bh * S_LEN * HDIM;

  auto stage = [&](int buf, int kk) {
#pragma unroll
    for (int rnd = 0; rnd < 4; ++rnd) {       // attn tile: 128x32 = 1024 x 16B
      int idx = rnd * 256 + tid;
      int r = idx >> 3, c4 = (idx & 7) * 4;
      cp16(&ab[(size_t)(m0 + r) * S_LEN + kk + c4], &sm.g.A[buf][r][c4]);
    }
#pragma unroll
    for (int rnd = 0; rnd < 2; ++rnd) {       // v tile: 32x64 = 512 x 16B
      int idx = rnd * 256 + tid;
      int r = idx >> 4, c4 = (idx & 15) * 4;
      cp16(&vb[(size_t)(kk + r) * HDIM + c4], &sm.g.B[buf][r][c4]);
    }
  };

  v8f acc[2][2] = {};
  stage(0, 0);
  async_wait();
  __syncthreads();

  int cur = 0;
  for (int kk = 0; kk < S_LEN; kk += 32) {
    if (kk + 32 < S_LEN) stage(cur ^ 1, kk + 32);
#pragma unroll
    for (int k0 = 0; k0 < 32; k0 += 4) {
      v2f a[2], b[2];
#pragma unroll
      for (int mi = 0; mi < 2; ++mi) {
        a[mi].x = sm.g.A[cur][WM + mi * 16 + lm][k0 + 2 * half];
        a[mi].y = sm.g.A[cur][WM + mi * 16 + lm][k0 + 2 * half + 1];
      }
#pragma unroll
      for (int ni = 0; ni < 2; ++ni) {
        b[ni].x = sm.g.B[cur][k0 + 2 * half][WN + ni * 16 + lm];
        b[ni].y = sm.g.B[cur][k0 + 2 * half + 1][WN + ni * 16 + lm];
      }
#pragma unroll
      for (int mi = 0; mi < 2; ++mi)
#pragma unroll
        for (int ni = 0; ni < 2; ++ni)
          acc[mi][ni] = wmma4(a[mi], b[ni], acc[mi][ni]);
    }
    async_wait();
    __syncthreads();
    cur ^= 1;
  }

  // relative-value epilogue: acc += W(128x65) @ emb_v(65x64) on this tile
  __syncthreads();
  for (int idx = tid; idx < 128 * NREL; idx += 256) {
    int r = idx / NREL, c = idx % NREL;
    sm.W[r][c] = wbuf[((size_t)bh * S_LEN + m0 + r) * NREL + c];
  }
  __syncthreads();

  for (int r = 0; r < NREL; ++r) {
#pragma unroll
    for (int ni = 0; ni < 2; ++ni) {
      const int d = WN + ni * 16 + lm;
      const float ev = emb_v[(size_t)r * HDIM + d];
#pragma unroll
      for (int mi = 0; mi < 2; ++mi)
#pragma unroll
        for (int vv = 0; vv < 8; ++vv)
          acc[mi][ni][vv] += sm.W[WM + mi * 16 + vv + 8 * half][r] * ev;
    }
  }

  const int b = bh >> 4, h = bh & 15;
#pragma unroll
  for (int mi = 0; mi < 2; ++mi)
#pragma unroll
    for (int ni = 0; ni < 2; ++ni) {
      const int d = WN + ni * 16 + lm;
#pragma unroll
      for (int vv = 0; vv < 8; ++vv) {
        int i = m0 + WM + mi * 16 + vv + 8 * half;
        ctx[(((size_t)b * S_LEN + i) * NHEAD + h) * HDIM + d] = acc[mi][ni][vv];
      }
    }
}

// ---------------------------------------------------------------------------
extern "C" void kernel_launch(void* const* d_in, const int* in_sizes, int n_in,
                              void* d_out, int out_size, void* d_ws,
                              size_t ws_size, hipStream_t stream) {
  (void)in_sizes; (void)n_in; (void)out_size; (void)ws_size;
  const float* key   = (const float*)d_in[0];
  const float* value = (const float*)d_in[1];
  const float* query = (const float*)d_in[2];
  const float* Wk = (const float*)d_in[3];
  const float* bk = (const float*)d_in[4];
  const float* Wq = (const float*)d_in[5];
  const float* bq = (const float*)d_in[6];
  const float* Wv = (const float*)d_in[7];
  const float* bv = (const float*)d_in[8];
  const float* Wo = (const float*)d_in[9];
  const float* bo = (const float*)d_in[10];
  const float* emb_k = (const float*)d_in[11];
  const float* emb_v = (const float*)d_in[12];

  float* out  = (float*)d_out;                       // [B,S,D]
  float* attn = out + (size_t)MROWS * D_MOD;         // [B,H,S,S]

  float* ws = (float*)d_ws;
  const size_t SZ_PROJ = (size_t)MROWS * D_MOD;      // 4,194,304 floats
  const size_t SZ_REL  = (size_t)BH * S_LEN * NREL;  // 4,259,840 floats
  float* qbuf  = ws;
  float* kbuf  = qbuf + SZ_PROJ;
  float* vbuf  = kbuf + SZ_PROJ;
  float* qebuf = vbuf + SZ_PROJ;
  float* wbuf  = qebuf + SZ_REL;
  float* ctxb  = wbuf + SZ_REL;                      // total ~101 MB

  dim3 blk(256);
  dim3 g_gemm(D_MOD / 128, MROWS / 64);
  const float qscale = 0.125f;  // 1/sqrt(DK)

  gemm_kernel<1><<<g_gemm, blk, 0, stream>>>(query, Wq, bq, qbuf, qscale);
  gemm_kernel<1><<<g_gemm, blk, 0, stream>>>(key,   Wk, bk, kbuf, 1.0f);
  gemm_kernel<1><<<g_gemm, blk, 0, stream>>>(value, Wv, bv, vbuf, 1.0f);

  qe_kernel<<<dim3(BH * S_LEN), dim3(96), 0, stream>>>(qbuf, emb_k, qebuf);

  scores_kernel<<<dim3(S_LEN / 128, S_LEN / 64, BH), blk, 0, stream>>>(
      qbuf, kbuf, qebuf, attn);

  softmax_kernel<<<dim3(BH * S_LEN / 8), blk, 0, stream>>>(attn, wbuf);

  ctx_kernel<<<dim3(S_LEN / 128, BH), blk, 0, stream>>>(attn, vbuf, wbuf,
                                                        emb_v, ctxb);

  gemm_kernel<0><<<g_gemm, blk, 0, stream>>>(ctxb, Wo, bo, out, 1.0f);
}